// SAGEGNN_76184129896626
// MI455X (gfx1250) — compile-verified
//
#include <hip/hip_runtime.h>

typedef float v2f __attribute__((ext_vector_type(2)));
typedef float v4f __attribute__((ext_vector_type(4)));
typedef float v8f __attribute__((ext_vector_type(8)));

#define N_NODES 8192
#define N_EDGES 262144
#define IN_DIM  128
#define HID     256

// ---------------- zero scratch ----------------
__global__ void k_zero(float* __restrict__ p, long n) {
    long i = (long)blockIdx.x * blockDim.x + threadIdx.x;
    long stride = (long)gridDim.x * blockDim.x;
    for (; i < n; i += stride) p[i] = 0.0f;
}

// ---------------- edge scatter: agg[dst] += x[src], deg[dst] += 1 ----------------
// one wave32 per edge; each lane owns dim/32 contiguous features (float4 chunks).
// agg/deg targets are small (<=8MB) and stay L2-resident -> L2 atomic units.
__global__ void k_scatter(const float* __restrict__ x, const long long* __restrict__ ei,
                          float* __restrict__ agg, float* __restrict__ deg,
                          int dim, int do_deg) {
    int wave = (int)((blockIdx.x * blockDim.x + threadIdx.x) >> 5);
    int lane = threadIdx.x & 31;
    if (wave >= N_EDGES) return;
    int src = (int)ei[wave];
    int dst = (int)ei[N_EDGES + wave];
    int per = dim >> 5;                       // 4 (dim=128) or 8 (dim=256)
    const float* xs = x   + (long)src * dim + lane * per;
    float*       ad = agg + (long)dst * dim + lane * per;
    for (int f = 0; f < per; f += 4) {
        float4 v = *(const float4*)(xs + f);
        atomicAdd(ad + f + 0, v.x);
        atomicAdd(ad + f + 1, v.y);
        atomicAdd(ad + f + 2, v.z);
        atomicAdd(ad + f + 3, v.w);
    }
    if (do_deg && lane == 0) atomicAdd(deg + dst, 1.0f);
}

// ---------------- fused SAGE layer GEMM (fp32 WMMA) ----------------
// out[8192][HID] = act( (agg/max(deg,1)) @ wl^T + bl + xin @ wr^T )
// one wave per 16x16 output tile; A/B per-lane layout per CDNA5 ISA 16x16x4 f32.
template <int K, bool RELU>
__global__ void k_sage_gemm(const float* __restrict__ agg, const float* __restrict__ deg,
                            const float* __restrict__ xin,
                            const float* __restrict__ wl, const float* __restrict__ bl,
                            const float* __restrict__ wr, float* __restrict__ out) {
    int wave  = (int)((blockIdx.x * blockDim.x + threadIdx.x) >> 5);
    int lane  = threadIdx.x & 31;
    int tileM = wave >> 4;                    // 512 row tiles
    int tileN = wave & 15;                    // 16 col tiles
    int row   = tileM * 16 + (lane & 15);     // A-matrix row for this lane (M = lane%16)
    int col   = tileN * 16 + (lane & 15);     // B-matrix col for this lane (N = lane%16)
    int kk    = (lane >> 4) * 2;              // K sub-offset: lanes16-31 hold K+2,K+3

    float scale = 1.0f / fmaxf(deg[row], 1.0f);
    v8f c = {};

    // pass 1: mean @ wl^T  (mean fused as agg * 1/deg)
    {
        const float* aB = agg + (long)row * K + kk;
        const float* bB = wl  + (long)col * K + kk;
        for (int k = 0; k < K; k += 4) {
            v2f a = *(const v2f*)(aB + k);
            a.x *= scale; a.y *= scale;
            v2f b = *(const v2f*)(bB + k);
            c = __builtin_amdgcn_wmma_f32_16x16x4_f32(false, a, false, b,
                                                      (short)0, c, false, false);
        }
    }
    // pass 2: accumulate xin @ wr^T into the same C tile
    {
        const float* aB = xin + (long)row * K + kk;
        const float* bB = wr  + (long)col * K + kk;
        for (int k = 0; k < K; k += 4) {
            v2f a = *(const v2f*)(aB + k);
            v2f b = *(const v2f*)(bB + k);
            c = __builtin_amdgcn_wmma_f32_16x16x4_f32(false, a, false, b,
                                                      (short)0, c, false, false);
        }
    }

    float bias  = bl[col];
    int   rbase = tileM * 16 + (lane >> 4) * 8;   // C layout: M = v + 8*(lane/16)
    for (int v = 0; v < 8; v++) {
        float val = c[v] + bias;
        if (RELU) val = fmaxf(val, 0.0f);
        out[(long)(rbase + v) * HID + col] = val;
    }
}

// ---------------- plain GEMM: out = xin @ w^T + b  (K = HID) ----------------
__global__ void k_gemm_out(const float* __restrict__ xin, const float* __restrict__ w,
                           const float* __restrict__ b, float* __restrict__ out) {
    int wave  = (int)((blockIdx.x * blockDim.x + threadIdx.x) >> 5);
    int lane  = threadIdx.x & 31;
    int tileM = wave >> 4;
    int tileN = wave & 15;
    int row   = tileM * 16 + (lane & 15);
    int col   = tileN * 16 + (lane & 15);
    int kk    = (lane >> 4) * 2;

    v8f c = {};
    const float* aB = xin + (long)row * HID + kk;
    const float* bB = w   + (long)col * HID + kk;
    for (int k = 0; k < HID; k += 4) {
        v2f a  = *(const v2f*)(aB + k);
        v2f bb = *(const v2f*)(bB + k);
        c = __builtin_amdgcn_wmma_f32_16x16x4_f32(false, a, false, bb,
                                                  (short)0, c, false, false);
    }
    float bias  = b[col];
    int   rbase = tileM * 16 + (lane >> 4) * 8;
    for (int v = 0; v < 8; v++)
        out[(long)(rbase + v) * HID + col] = c[v] + bias;
}

// ---------------- li = h @ w_left, lj = h @ w_right (wave per node) ----------------
__global__ void k_lilj(const float* __restrict__ h, const float* __restrict__ w_edge,
                       float* __restrict__ li, float* __restrict__ lj) {
    int wave = (int)((blockIdx.x * blockDim.x + threadIdx.x) >> 5);
    int lane = threadIdx.x & 31;
    if (wave >= N_NODES) return;
    const float* hr = h + (long)wave * HID;
    float sl = 0.0f, sr = 0.0f;
    int base = lane * 8;                       // 8 features per lane
    for (int f = base; f < base + 8; f += 4) {
        float4 v = *(const float4*)(hr + f);
        float4 a = *(const float4*)(w_edge + f);
        float4 b = *(const float4*)(w_edge + HID + f);
        sl += v.x * a.x + v.y * a.y + v.z * a.z + v.w * a.w;
        sr += v.x * b.x + v.y * b.y + v.z * b.z + v.w * b.w;
    }
    for (int m = 16; m > 0; m >>= 1) {
        sl += __shfl_xor(sl, m, 32);
        sr += __shfl_xor(sr, m, 32);
    }
    if (lane == 0) { li[wave] = sl; lj[wave] = sr; }
}

// ---------------- out[i][j] = li[i] + lj[j] + b_edge ----------------
// 256 MB one-way stream > 192 MB L2: use non-temporal (TH=NT) 128-bit stores so
// the stream doesn't rinse/evict L2 and write-allocate traffic is avoided.
// NT builtin requires a clang ext-vector type (v4f), not HIP's float4 struct.
__global__ void k_outer(const float* __restrict__ li, const float* __restrict__ lj,
                        const float* __restrict__ b_edge, float* __restrict__ out) {
    long idx = (long)blockIdx.x * blockDim.x + threadIdx.x;   // one v4f per thread
    int  i   = (int)(idx >> 11);                               // 2048 float4 per row
    int  j4  = (int)(idx & 2047) << 2;
    float a  = li[i] + b_edge[0];
    v4f  v   = *(const v4f*)(lj + j4);                         // 32 KB, L2-resident (RT)
    v4f  r   = {a + v.x, a + v.y, a + v.z, a + v.w};
    __builtin_nontemporal_store(r, (v4f*)(out + ((long)i << 13) + j4));
}

extern "C" void kernel_launch(void* const* d_in, const int* in_sizes, int n_in,
                              void* d_out, int out_size, void* d_ws, size_t ws_size,
                              hipStream_t stream) {
    const float*     x      = (const float*)d_in[0];
    const long long* ei     = (const long long*)d_in[1];   // int64 edge_index (2, N_EDGES)
    const float*     wl1    = (const float*)d_in[2];
    const float*     bl1    = (const float*)d_in[3];
    const float*     wr1    = (const float*)d_in[4];
    const float*     wl2    = (const float*)d_in[5];
    const float*     bl2    = (const float*)d_in[6];
    const float*     wr2    = (const float*)d_in[7];
    const float*     w_out  = (const float*)d_in[8];
    const float*     b_out  = (const float*)d_in[9];
    const float*     w_edge = (const float*)d_in[10];
    const float*     b_edge = (const float*)d_in[11];
    float*           out    = (float*)d_out;

    // workspace layout (floats), ~38 MB total
    float* ws   = (float*)d_ws;
    float* agg1 = ws;                                   // 8192*128
    float* deg  = agg1 + (long)N_NODES * IN_DIM;        // 8192
    float* agg2 = deg  + N_NODES;                       // 8192*256
    float* h1   = agg2 + (long)N_NODES * HID;           // 8192*256
    float* h2   = h1   + (long)N_NODES * HID;           // 8192*256
    float* hh   = h2   + (long)N_NODES * HID;           // 8192*256
    float* li   = hh   + (long)N_NODES * HID;           // 8192
    float* lj   = li   + N_NODES;                       // 8192

    // 1. zero accumulators (agg1 + deg + agg2 are contiguous)
    long nz = (long)N_NODES * (IN_DIM + 1 + HID);
    k_zero<<<2048, 256, 0, stream>>>(agg1, nz);

    // 2. layer-1 scatter (also computes degrees, reused for layer 2)
    k_scatter<<<(N_EDGES * 32) / 256, 256, 0, stream>>>(x, ei, agg1, deg, IN_DIM, 1);

    // 3. layer-1 fused GEMM: h1 = relu(mean@wl1^T + bl1 + x@wr1^T)   [8192 waves]
    k_sage_gemm<IN_DIM, true><<<(512 * 16 * 32) / 256, 256, 0, stream>>>(
        agg1, deg, x, wl1, bl1, wr1, h1);

    // 4. layer-2 scatter on h1
    k_scatter<<<(N_EDGES * 32) / 256, 256, 0, stream>>>(h1, ei, agg2, deg, HID, 0);

    // 5. layer-2 fused GEMM: h2 = relu(mean2@wl2^T + bl2 + h1@wr2^T)
    k_sage_gemm<HID, true><<<(512 * 16 * 32) / 256, 256, 0, stream>>>(
        agg2, deg, h1, wl2, bl2, wr2, h2);

    // 6. output projection: hh = h2 @ w_out^T + b_out
    k_gemm_out<<<(512 * 16 * 32) / 256, 256, 0, stream>>>(h2, w_out, b_out, hh);

    // 7. edge score halves
    k_lilj<<<(N_NODES * 32) / 256, 256, 0, stream>>>(hh, w_edge, li, lj);

    // 8. broadcast-add outer sum (256 MB NT streaming write — the roofline term)
    k_outer<<<(long)N_NODES * (N_NODES / 4) / 256, 256, 0, stream>>>(li, lj, b_edge, out);
}